// GCN_class_64295660421704
// MI455X (gfx1250) — compile-verified
//
#include <hip/hip_runtime.h>
#include <hip/hip_bf16.h>

#define NN 100000
#define NE 1600000
#define NG 256
#define NF 32
#define NH 64

typedef __attribute__((ext_vector_type(2))) float v2f;
typedef __attribute__((ext_vector_type(8))) float v8f;

// ---------------- utility ----------------
__global__ void k_fill0(float* __restrict__ p, int n) {
    int i = blockIdx.x * blockDim.x + threadIdx.x;
    if (i < n) p[i] = 0.0f;
}

// degree over dst (one atomic per edge)
__global__ void k_degree(const int* __restrict__ ei, float* __restrict__ deg) {
    int e = blockIdx.x * blockDim.x + threadIdx.x;
    if (e < NE) unsafeAtomicAdd(&deg[ei[NE + e]], 1.0f);
}

// per-graph node counts
__global__ void k_counts(const int* __restrict__ batch, float* __restrict__ cnt) {
    int i = blockIdx.x * blockDim.x + threadIdx.x;
    if (i < NN) unsafeAtomicAdd(&cnt[batch[i]], 1.0f);
}

__global__ void k_dinv(const float* __restrict__ deg, float* __restrict__ dinv) {
    int i = blockIdx.x * blockDim.x + threadIdx.x;
    if (i < NN) dinv[i] = rsqrtf(deg[i] + 1.0f);
}

// ---------------- WMMA GEMM: H[M x 64] = X[M x K] @ W[K x 64] ----------------
// One 16x16 output tile per wave; V_WMMA_F32_16X16X4_F32, K-loop step 4.
__global__ __launch_bounds__(256) void k_gemm_wmma(const float* __restrict__ X,
                                                   const float* __restrict__ W,
                                                   float* __restrict__ H,
                                                   int M, int K) {
    __shared__ float sW[NH * NH];           // up to 64x64 = 16 KB
    const int tid = threadIdx.x;
    for (int i = tid; i < K * NH; i += 256) sW[i] = W[i];
    __syncthreads();

    const int wave   = tid >> 5;
    const int lane   = tid & 31;
    const int tileId = blockIdx.x * 8 + wave;
    const int nTiles = (M >> 4) * (NH >> 4);
    if (tileId >= nTiles) return;           // uniform per wave

    const int tn  = tileId & 3;             // NH/16 == 4
    const int tm  = tileId >> 2;
    const int row = lane & 15;              // A: M index / B,C: N index
    const int kh  = (lane >> 4) << 1;       // lane half -> K sub-offset {0,2}

    const float* xrow = X + (size_t)(tm * 16 + row) * K;

    v8f c = {};
    for (int k0 = 0; k0 < K; k0 += 4) {
        v2f a, b;
        // A 16x4 f32: lanes 0-15 hold K={k0,k0+1}, lanes 16-31 hold K={k0+2,k0+3}
        a.x = xrow[k0 + kh];
        a.y = xrow[k0 + kh + 1];
        // B 4x16 f32: row striped across lanes, lane halves cover K pairs
        b.x = sW[(k0 + kh) * NH + tn * 16 + row];
        b.y = sW[(k0 + kh + 1) * NH + tn * 16 + row];
        c = __builtin_amdgcn_wmma_f32_16x16x4_f32(false, a, false, b,
                                                  (short)0, c, false, false);
    }

    // C/D 16x16 f32: VGPR r -> M = r + 8*(lane>=16), N = lane&15
    float* hout = H + (size_t)(tm * 16) * NH + tn * 16;
    const int mo = (lane >> 4) << 3;
    const int n  = lane & 15;
#pragma unroll
    for (int r = 0; r < 8; ++r)
        hout[(size_t)(r + mo) * NH + n] = c[r];
}

// ---------------- edge aggregate: agg[dst] += h[src] * dinv[src]*dinv[dst] ----
// one (edge, feature) pair per thread: 64 lanes cover 64 features of one edge
__global__ void k_aggregate(const int* __restrict__ ei,
                            const float* __restrict__ h,
                            const float* __restrict__ dinv,
                            float* __restrict__ agg) {
    int t = blockIdx.x * blockDim.x + threadIdx.x;   // < 1.6M*64 = 102.4M
    int e = t >> 6;
    int f = t & 63;
    if (e >= NE) return;
    int s = ei[e];
    int d = ei[NE + e];
    float nrm = dinv[s] * dinv[d];
    float v   = h[(size_t)s * NH + f] * nrm;
    unsafeAtomicAdd(&agg[(size_t)d * NH + f], v);
}

// layer-1 epilogue: self-loop + bias + ReLU
__global__ void k_finish_relu(const float* __restrict__ agg,
                              const float* __restrict__ h,
                              const float* __restrict__ dinv,
                              const float* __restrict__ b,
                              float* __restrict__ out) {
    int t = blockIdx.x * blockDim.x + threadIdx.x;   // < 100000*64
    if (t >= NN * NH) return;
    int i = t >> 6, f = t & 63;
    float di = dinv[i];
    float v = agg[t] + h[t] * (di * di) + b[f];
    out[t] = fmaxf(v, 0.0f);
}

// layer-2 epilogue fused with mean-pool numerator (atomic scatter by graph id)
__global__ void k_finish_pool(const float* __restrict__ agg,
                              const float* __restrict__ h,
                              const float* __restrict__ dinv,
                              const float* __restrict__ b,
                              const int* __restrict__ batch,
                              float* __restrict__ pooled) {
    int t = blockIdx.x * blockDim.x + threadIdx.x;
    if (t >= NN * NH) return;
    int i = t >> 6, f = t & 63;
    float di = dinv[i];
    float v = agg[t] + h[t] * (di * di) + b[f];
    unsafeAtomicAdd(&pooled[(size_t)batch[i] * NH + f], v);
}

// final: out[g,:2] = (pooled[g]/max(cnt,1)) @ Wl + bl   (one thread per graph)
__global__ void k_final(const float* __restrict__ pooled,
                        const float* __restrict__ cnt,
                        const float* __restrict__ Wl,
                        const float* __restrict__ bl,
                        float* __restrict__ out) {
    int g = threadIdx.x;                     // 256 threads, 1 block
    if (g >= NG) return;
    float inv = 1.0f / fmaxf(cnt[g], 1.0f);
    float s0 = 0.0f, s1 = 0.0f;
    for (int f = 0; f < NH; ++f) {
        float v = pooled[g * NH + f] * inv;
        s0 += v * Wl[f * 2 + 0];
        s1 += v * Wl[f * 2 + 1];
    }
    out[g * 2 + 0] = s0 + bl[0];
    out[g * 2 + 1] = s1 + bl[1];
}

extern "C" void kernel_launch(void* const* d_in, const int* in_sizes, int n_in,
                              void* d_out, int out_size, void* d_ws, size_t ws_size,
                              hipStream_t stream) {
    const float* x  = (const float*)d_in[0];
    const int*   ei = (const int*)d_in[1];
    const int*   bt = (const int*)d_in[2];
    const float* W1 = (const float*)d_in[3];
    const float* b1 = (const float*)d_in[4];
    const float* W2 = (const float*)d_in[5];
    const float* b2 = (const float*)d_in[6];
    const float* Wl = (const float*)d_in[7];
    const float* bl = (const float*)d_in[8];
    float* out = (float*)d_out;

    // workspace layout (floats)
    float* ws     = (float*)d_ws;
    float* deg    = ws;                       // 100000
    float* dinv   = deg  + NN;                // 100000
    float* cnt    = dinv + NN;                // 256
    float* pooled = cnt  + NG;                // 16384
    float* bufA   = pooled + NG * NH;         // 6.4M  (h1 / h2)
    float* bufB   = bufA + (size_t)NN * NH;   // 6.4M  (agg, zeroed per use)
    float* bufC   = bufB + (size_t)NN * NH;   // 6.4M  (relu(h1))

    const int T = 256;
    const int headN = 2 * NN + NG + NG * NH;      // deg,dinv,cnt,pooled
    const int aggN  = NN * NH;
    const int nodeT = NN * NH;                    // per-(node,feature) threads
    const int edgeT = NE * 64;                    // per-(edge,feature) threads

    // zero deg/dinv/cnt/pooled and agg buffer
    k_fill0<<<(headN + T - 1) / T, T, 0, stream>>>(ws, headN);
    k_fill0<<<(aggN + T - 1) / T, T, 0, stream>>>(bufB, aggN);

    k_degree<<<(NE + T - 1) / T, T, 0, stream>>>(ei, deg);
    k_counts<<<(NN + T - 1) / T, T, 0, stream>>>(bt, cnt);
    k_dinv  <<<(NN + T - 1) / T, T, 0, stream>>>(deg, dinv);

    // layer 1: h1 = x @ W1  (WMMA), aggregate, epilogue+ReLU
    k_gemm_wmma<<<(NN / 16) * (NH / 16) / 8, T, 0, stream>>>(x, W1, bufA, NN, NF);
    k_aggregate<<<(edgeT + T - 1) / T, T, 0, stream>>>(ei, bufA, dinv, bufB);
    k_finish_relu<<<(nodeT + T - 1) / T, T, 0, stream>>>(bufB, bufA, dinv, b1, bufC);

    // layer 2: h2 = relu(h1) @ W2 (WMMA), aggregate, epilogue fused with pooling
    k_gemm_wmma<<<(NN / 16) * (NH / 16) / 8, T, 0, stream>>>(bufC, W2, bufA, NN, NH);
    k_fill0<<<(aggN + T - 1) / T, T, 0, stream>>>(bufB, aggN);
    k_aggregate<<<(edgeT + T - 1) / T, T, 0, stream>>>(ei, bufA, dinv, bufB);
    k_finish_pool<<<(nodeT + T - 1) / T, T, 0, stream>>>(bufB, bufA, dinv, b2, bt, pooled);

    // final classifier
    k_final<<<1, T, 0, stream>>>(pooled, cnt, Wl, bl, out);
}